// MaxsatLayer_12979391169094
// MI455X (gfx1250) — compile-verified
//
#include <hip/hip_runtime.h>
#include <math.h>

// Problem constants (match reference)
#define N_  24
#define M_  128
#define K_  64
#define MAX_ITER 8
#define EPS_F 1e-8f
#define PI_F 3.14159265358979323846f

typedef __attribute__((ext_vector_type(2))) float v2f;
typedef __attribute__((ext_vector_type(8))) float v8f;

// ---------------------------------------------------------------------------
// Kernel 1: G = S * S^T  (24x24, diagonal zeroed), via V_WMMA_F32_16X16X4_F32.
// One wave computes the 2x2 grid of 16x16 tiles (rows/cols >= 24 are padded
// with zeros on load and masked on store). K=128 accumulated in steps of 4.
// ---------------------------------------------------------------------------
__global__ __launch_bounds__(32) void gram_wmma_kernel(
    const float* __restrict__ S, float* __restrict__ G) {
  const int lane = threadIdx.x & 31;
  const int half = lane >> 4;   // K-pair selector for A/B striping
  const int m16  = lane & 15;   // row (A) / col (B) within tile

  #pragma unroll
  for (int t = 0; t < 4; ++t) {
    const int it = t >> 1, jt = t & 1;
    const int arow = it * 16 + m16;     // A = S tile (16 x 4 slice of K)
    const int brow = jt * 16 + m16;     // B = S^T tile (4 x 16) -> row of S
    const bool av = (arow < N_);
    const bool bv = (brow < N_);
    v8f acc = {};
    #pragma unroll
    for (int k0 = 0; k0 < M_; k0 += 4) {
      const int ka = k0 + half * 2;
      v2f a, b;
      a.x = av ? S[arow * M_ + ka]     : 0.0f;
      a.y = av ? S[arow * M_ + ka + 1] : 0.0f;
      b.x = bv ? S[brow * M_ + ka]     : 0.0f;
      b.y = bv ? S[brow * M_ + ka + 1] : 0.0f;
      // D = A(16x4) * B(4x16) + C   -> v_wmma_f32_16x16x4_f32
      acc = __builtin_amdgcn_wmma_f32_16x16x4_f32(
          false, a, false, b, (short)0, acc, false, false);
    }
    // C/D layout: vgpr r, lane L -> (row = r + 8*(L>=16), col = L&15)
    #pragma unroll
    for (int r = 0; r < 8; ++r) {
      const int row = it * 16 + r + half * 8;
      const int col = jt * 16 + m16;
      float val = acc[r];
      if (row == col) val = 0.0f;            // fold out the snorm*v_i term
      if (row < N_ && col < N_) G[row * N_ + col] = val;
    }
  }
}

// ---------------------------------------------------------------------------
// Cross-lane (wave32) sum via butterfly shuffles (lowers to ds/permlane ops).
// ---------------------------------------------------------------------------
__device__ __forceinline__ float wave_sum(float v) {
  #pragma unroll
  for (int off = 16; off > 0; off >>= 1)
    v += __shfl_xor(v, off, 32);
  return v;
}

// ---------------------------------------------------------------------------
// Kernel 2: one wave32 per batch element. Lane L owns columns k=L and k=L+32
// of V (24x64) -> V lives entirely in 48 VGPRs per lane. G is broadcast from
// LDS. 8 sweeps x 24 Gauss-Seidel steps, fully unrolled over rows so V stays
// register-resident (no dynamic VGPR indexing).
// ---------------------------------------------------------------------------
__global__ __launch_bounds__(256) void mix_solver_kernel(
    const float* __restrict__ z, const float* __restrict__ Vr,
    const int* __restrict__ is_input, const float* __restrict__ G,
    float* __restrict__ out, int B) {
  __shared__ float Gs[N_ * N_];
  for (int idx = threadIdx.x; idx < N_ * N_; idx += blockDim.x)
    Gs[idx] = G[idx];
  __syncthreads();

  const int lane = threadIdx.x & 31;
  const int wid  = threadIdx.x >> 5;
  const int b = blockIdx.x * (blockDim.x >> 5) + wid;
  if (b >= B) return;

  const float* vr = Vr + (size_t)b * (N_ * K_);

  // ---- load + row-normalize: r = normalize(V_rand[b]) ----
  float vlo[N_], vhi[N_];
  #pragma unroll
  for (int j = 0; j < N_; ++j) {
    float lo = vr[j * K_ + lane];
    float hi = vr[j * K_ + 32 + lane];
    float s  = wave_sum(lo * lo + hi * hi);
    float inv = 1.0f / (sqrtf(s) + EPS_F);
    vlo[j] = lo * inv;
    vhi[j] = hi * inv;
  }
  const float v0lo = vlo[0], v0hi = vhi[0];

  // ---- per-wave keep mask from is_input (uniform across lanes) ----
  int flag = (lane < N_) ? is_input[(size_t)b * N_ + lane] : 0;
  const unsigned keepmask = (unsigned)__ballot(flag > 0);

  // ---- build initial V: phase-encoded rows where is_input, else r ----
  #pragma unroll
  for (int j = 0; j < N_; ++j) {
    if ((keepmask >> j) & 1u) {
      float proj = wave_sum(vlo[j] * v0lo + vhi[j] * v0hi);
      float ulo = vlo[j] - proj * v0lo;
      float uhi = vhi[j] - proj * v0hi;
      float un  = wave_sum(ulo * ulo + uhi * uhi);
      float uinv = 1.0f / (sqrtf(un) + EPS_F);
      ulo *= uinv; uhi *= uinv;
      float zj = z[(size_t)b * N_ + j];
      float cz = __cosf(PI_F * zj);
      float sz = __sinf(PI_F * zj);
      vlo[j] = sz * ulo - cz * v0lo;
      vhi[j] = sz * uhi - cz * v0hi;
    }
  }
  vlo[0] = v0lo; vhi[0] = v0hi;

  // ---- 8 sweeps of Gauss-Seidel: g_i = sum_{j!=i} G[i,j] * V[j] ----
  for (int it = 0; it < MAX_ITER; ++it) {
    #pragma unroll
    for (int i = 0; i < N_; ++i) {
      if ((keepmask >> i) & 1u) continue;   // input rows never move
      const float* gr = &Gs[i * N_];        // uniform LDS row (diag is 0)
      float a0 = 0.f, a1 = 0.f, b0 = 0.f, b1 = 0.f;  // 2 chains: hide FMA latency
      #pragma unroll
      for (int j = 0; j < N_; j += 2) {
        float g0 = gr[j], g1 = gr[j + 1];
        a0 = fmaf(g0, vlo[j], a0);     b0 = fmaf(g0, vhi[j], b0);
        a1 = fmaf(g1, vlo[j + 1], a1); b1 = fmaf(g1, vhi[j + 1], b1);
      }
      float glo = a0 + a1, ghi = b0 + b1;
      float nrm2 = wave_sum(glo * glo + ghi * ghi);
      float sc = -1.0f / (sqrtf(nrm2) + EPS_F);
      vlo[i] = glo * sc;
      vhi[i] = ghi * sc;
    }
  }

  // ---- epilogue: out[b,j] = arccos(clip(-V[j].v0)) / pi ----
  float myout = 0.0f;
  #pragma unroll
  for (int j = 0; j < N_; ++j) {
    float d = wave_sum(vlo[j] * v0lo + vhi[j] * v0hi);
    float c = fminf(fmaxf(-d, -1.0f + 1e-6f), 1.0f - 1e-6f);
    float o = acosf(c) * (1.0f / PI_F);
    if (lane == j) myout = o;
  }
  if (lane < N_) out[(size_t)b * N_ + lane] = myout;
}

// ---------------------------------------------------------------------------
extern "C" void kernel_launch(void* const* d_in, const int* in_sizes, int n_in,
                              void* d_out, int out_size, void* d_ws, size_t ws_size,
                              hipStream_t stream) {
  const float* z        = (const float*)d_in[0];   // (B, N)
  const float* S        = (const float*)d_in[1];   // (N, M)
  const float* Vr       = (const float*)d_in[2];   // (B, N, K)
  const int*   is_input = (const int*)d_in[3];     // (B, N)
  float* out = (float*)d_out;                      // (B, N)
  float* G   = (float*)d_ws;                       // 24*24 floats of scratch

  const int B = in_sizes[0] / N_;

  gram_wmma_kernel<<<1, 32, 0, stream>>>(S, G);

  const int wavesPerBlock = 8;                     // 256 threads = 8 wave32
  const int blocks = (B + wavesPerBlock - 1) / wavesPerBlock;
  mix_solver_kernel<<<blocks, wavesPerBlock * 32, 0, stream>>>(
      z, Vr, is_input, G, out, B);
}